// NEUMF_Feature_inner_product_Sample_40699110097051
// MI455X (gfx1250) — compile-verified
//
#include <hip/hip_runtime.h>
#include <math.h>

typedef __attribute__((ext_vector_type(2))) float v2f;
typedef __attribute__((ext_vector_type(4))) float v4f;
typedef __attribute__((ext_vector_type(8))) float v8f;

#define N_NODES  100000
#define IN_DIM   512
#define OUT_DIM  128
#define N_EDGES  300000

// LDS row stride in floats: 512 + 4 padding. Stride == 4 (mod 64) means the
// b64 K-pair reads (lanes 0-15 at +0, lanes 16-31 at +2 dwords) touch all 64
// banks exactly once per ds_load -> conflict-free.
#define LDS_STRIDE (IN_DIM + 4)
#define M_TILE     32                      // rows staged per block (2 WMMA M tiles)
#define LDS_BYTES  (M_TILE * LDS_STRIDE * 4)  // 66,048 B -> dynamic LDS (320KB/WGP)

// ---------------------------------------------------------------------------
// Kernel 1: z = x @ weight via V_WMMA_F32_16X16X4_F32 (fp32 matrix pipe).
// Block = 256 threads = 8 waves; block computes a 32(M) x 128(N) tile of z.
// Wave w owns columns [16w, 16w+16) and BOTH M tiles, so each B fragment
// (global load) feeds two WMMAs -> half the weight-cache traffic per FLOP.
// ---------------------------------------------------------------------------
__global__ __launch_bounds__(256) void encode_gemm_wmma(
    const float* __restrict__ x, const float* __restrict__ w,
    float* __restrict__ z) {
  extern __shared__ float As[];  // [M_TILE][LDS_STRIDE]

  const int m0  = blockIdx.x * M_TILE;
  const int tid = threadIdx.x;

  // --- stage the 32x512 A tile into LDS, coalesced float4 loads ---
  {
    const int row = tid >> 3;  // 0..31
    const int c0  = tid & 7;   // 0..7
    const float* src = x + (size_t)(m0 + row) * IN_DIM;
    float* dst = As + row * LDS_STRIDE;
#pragma unroll
    for (int j = 0; j < 16; ++j) {
      const int c4 = (c0 + 8 * j) * 4;  // 8 lanes x float4 = contiguous 128B
      *(v4f*)(dst + c4) = *(const v4f*)(src + c4);
    }
  }
  __syncthreads();

  const int wave = tid >> 5;
  const int lane = tid & 31;
  const int half = lane >> 4;   // 0: lanes 0-15 (K0/K1), 1: lanes 16-31 (K2/K3)
  const int l16  = lane & 15;
  const int ncol = wave * 16 + l16;

  const float* arow0 = As + l16 * LDS_STRIDE;         // M tile 0
  const float* arow1 = As + (16 + l16) * LDS_STRIDE;  // M tile 1

  v8f c0 = {};  // accumulator for rows [m0, m0+16)
  v8f c1 = {};  // accumulator for rows [m0+16, m0+32)

#pragma unroll 4
  for (int kb = 0; kb < IN_DIM / 4; ++kb) {
    const int kk = kb * 4 + half * 2;
    // B fragment (4x16): v0 = K0/K2 row, v1 = K1/K3 row; shared by both tiles.
    v2f b;
    b.x = w[(size_t)(kk + 0) * OUT_DIM + ncol];
    b.y = w[(size_t)(kk + 1) * OUT_DIM + ncol];
    // A fragments (16x4, 32-bit): v0 = K0 (lanes 0-15)/K2 (16-31), v1 = K1/K3.
    v2f a0 = *(const v2f*)(arow0 + kk);
    v2f a1 = *(const v2f*)(arow1 + kk);
    c0 = __builtin_amdgcn_wmma_f32_16x16x4_f32(false, a0, false, b,
                                               (short)0, c0, false, false);
    c1 = __builtin_amdgcn_wmma_f32_16x16x4_f32(false, a1, false, b,
                                               (short)0, c1, false, false);
  }

  // C/D layout: VGPR r -> lanes 0-15: M=r, lanes 16-31: M=8+r; N = lane%16.
#pragma unroll
  for (int r = 0; r < 8; ++r) {
    const int m = m0 + half * 8 + r;
    z[(size_t)m * OUT_DIM + ncol] = c0[r];
  }
#pragma unroll
  for (int r = 0; r < 8; ++r) {
    const int m = m0 + 16 + half * 8 + r;
    z[(size_t)m * OUT_DIM + ncol] = c1[r];
  }
}

// ---------------------------------------------------------------------------
// Kernel 2: edge decode + sigmoid. One wave32 per edge; z is L2-resident
// (51.2 MB << 192 MB L2), so the random gathers are served at L2 bandwidth.
// out[e] = sigmoid( sum_d z[a][d] * z[b][d] * w3[d] )
// ---------------------------------------------------------------------------
__global__ __launch_bounds__(256) void decode_edges(
    const float* __restrict__ z, const int* __restrict__ e_true,
    const int* __restrict__ e_false, const float* __restrict__ w3,
    float* __restrict__ out) {
  const int wave = threadIdx.x >> 5;
  const int lane = threadIdx.x & 31;
  const int e = blockIdx.x * 8 + wave;  // whole wave handles one edge (uniform)
  if (e >= 2 * N_EDGES) return;

  const int* edges = (e < N_EDGES) ? e_true : e_false;
  const int ee     = (e < N_EDGES) ? e : (e - N_EDGES);
  const int na = edges[2 * ee + 0];
  const int nb = edges[2 * ee + 1];

  // 128 dims / 32 lanes = one float4 per lane; 32 lanes read contiguous 512B.
  const v4f za = *(const v4f*)(z + (size_t)na * OUT_DIM + lane * 4);
  const v4f zb = *(const v4f*)(z + (size_t)nb * OUT_DIM + lane * 4);
  const v4f w4 = *(const v4f*)(w3 + lane * 4);

  float s = za.x * zb.x * w4.x + za.y * zb.y * w4.y +
            za.z * zb.z * w4.z + za.w * zb.w * w4.w;

  // wave32 butterfly reduction
#pragma unroll
  for (int off = 16; off > 0; off >>= 1) s += __shfl_xor(s, off, 32);

  if (lane == 0) out[e] = 1.0f / (1.0f + __expf(-s));
}

// ---------------------------------------------------------------------------
extern "C" void kernel_launch(void* const* d_in, const int* in_sizes, int n_in,
                              void* d_out, int out_size, void* d_ws,
                              size_t ws_size, hipStream_t stream) {
  const float* x   = (const float*)d_in[0];  // [100000, 512] f32
  const int*   e_t = (const int*)d_in[1];    // [300000, 2] int32
  const int*   e_f = (const int*)d_in[2];    // [300000, 2] int32
  const float* w   = (const float*)d_in[3];  // [512, 128] f32
  const float* w3  = (const float*)d_in[4];  // [128, 1] f32
  float* out = (float*)d_out;                // [600000, 1] f32

  float* z = (float*)d_ws;  // [100000, 128] f32 = 51.2 MB scratch

  encode_gemm_wmma<<<N_NODES / M_TILE, 256, LDS_BYTES, stream>>>(x, w, z);
  decode_edges<<<(2 * N_EDGES) / 8, 256, 0, stream>>>(z, e_t, e_f, w3, out);
}